// PointCloudAE_67585605370267
// MI455X (gfx1250) — compile-verified
//
#include <hip/hip_runtime.h>

typedef __attribute__((ext_vector_type(2))) float v2f;
typedef __attribute__((ext_vector_type(8))) float v8f;

#define B_SZ   16
#define N_IN   4096
#define N_OUT  2048
#define EMB_OFF   16777216            // 16*256*4096
#define OUTPC_OFF (EMB_OFF + 16*256)  // + emb
#define LOSS_OFF  (OUTPC_OFF + 16*2048*3)

// ---------------------------------------------------------------- init ws
__global__ void k_init(int* __restrict__ gfeat, int* __restrict__ maxfeat,
                       int* __restrict__ minq) {
  int i = blockIdx.x * 256 + threadIdx.x;
  if (i < 2048)  gfeat[i]   = 0;           // ReLU>=0 -> 0.0f bits
  if (i < 16384) maxfeat[i] = 0;
  if (i < 32768) minq[i]    = 0x7F7FFFFF;  // FLT_MAX bits
}

// -------------------------------------------- encoder conv1+conv2 -> gfeat
__global__ __launch_bounds__(256) void k_enc12(
    const float* __restrict__ pc,
    const float* __restrict__ w1, const float* __restrict__ b1,
    const float* __restrict__ w2, const float* __restrict__ b2,
    int* __restrict__ gfeat) {
  __shared__ int sg[128];
  const int tid = threadIdx.x;
  if (tid < 128) sg[tid] = 0;
  __syncthreads();
  const int b = blockIdx.x >> 4;
  const int n = ((blockIdx.x & 15) << 8) + tid;
  const float* pt = pc + ((size_t)b * N_IN + n) * 3;
  const float x = pt[0], y = pt[1], z = pt[2];
  float h[64];
#pragma unroll
  for (int j = 0; j < 64; ++j)
    h[j] = fmaxf(w1[j*3]*x + w1[j*3+1]*y + w1[j*3+2]*z + b1[j], 0.f);
  for (int o = 0; o < 128; ++o) {
    float a = b2[o];
    const float* wr = w2 + o * 64;
#pragma unroll
    for (int j = 0; j < 64; ++j) a += wr[j] * h[j];
    a = fmaxf(a, 0.f);
    atomicMax(&sg[o], __float_as_int(a));   // nonneg: int cmp == float cmp
  }
  __syncthreads();
  if (tid < 128) atomicMax(&gfeat[b*128 + tid], sg[tid]);
}

// ---- conv3 GEMM (WMMA f32), writes x0 to d_out, fused max-pool -> maxfeat
__global__ __launch_bounds__(256) void k_conv3(
    const float* __restrict__ pc,
    const float* __restrict__ w1, const float* __restrict__ b1,
    const float* __restrict__ w2, const float* __restrict__ b2,
    const float* __restrict__ w3, const float* __restrict__ b3,
    const int* __restrict__ gfeat,
    float* __restrict__ x0out, int* __restrict__ maxfeat) {
  __shared__ float x0s[256][16];
  const int b  = blockIdx.y;
  const int n0 = blockIdx.x << 4;
  const int tid = threadIdx.x;
  // --- build the 256x16 x0 tile (recompute conv1/conv2 for 16 points) ---
  {
    const int p = tid & 15, cg = tid >> 4;
    const float* pt = pc + ((size_t)b * N_IN + n0 + p) * 3;
    const float x = pt[0], y = pt[1], z = pt[2];
    float h[64];
#pragma unroll
    for (int j = 0; j < 64; ++j)
      h[j] = fmaxf(w1[j*3]*x + w1[j*3+1]*y + w1[j*3+2]*z + b1[j], 0.f);
    for (int i = 0; i < 8; ++i) {
      const int c = cg * 8 + i;
      float a = b2[c];
      const float* wr = w2 + c * 64;
#pragma unroll
      for (int j = 0; j < 64; ++j) a += wr[j] * h[j];
      a = fmaxf(a, 0.f);
      x0s[c][p] = a;
      x0out[((size_t)b*256 + c) * N_IN + n0 + p] = a;
    }
#pragma unroll
    for (int i = 0; i < 8; ++i) {          // gfeat broadcast rows 128..255
      const int idx = tid * 8 + i;
      const int c = idx >> 4, pp = idx & 15;
      const float gv = __int_as_float(gfeat[b*128 + c]);
      x0s[128 + c][pp] = gv;
      x0out[((size_t)b*256 + 128 + c) * N_IN + n0 + pp] = gv;
    }
  }
  __syncthreads();
  // --- 1024x16 = W3(1024x256) * x0tile(256x16) via v_wmma_f32_16x16x4_f32
  const int lane = tid & 31, wave = tid >> 5;
  const int lo = lane & 15, hi = lane >> 4, hi2 = hi << 1;
  const float* W3w = w3 + (size_t)(wave * 128) * 256;
  v8f acc[8] = {};
  for (int kk = 0; kk < 256; kk += 4) {
    v2f bf;
    bf.x = x0s[kk + hi2][lo];
    bf.y = x0s[kk + hi2 + 1][lo];
#pragma unroll
    for (int t = 0; t < 8; ++t) {
      v2f af = *(const v2f*)(W3w + (size_t)(t*16 + lo) * 256 + kk + hi2);
      acc[t] = __builtin_amdgcn_wmma_f32_16x16x4_f32(
          false, af, false, bf, (short)0, acc[t], false, false);
    }
  }
  // --- bias + relu + max over the 16 points + global atomic max ---
#pragma unroll
  for (int t = 0; t < 8; ++t) {
    const int rbase = wave * 128 + t * 16 + hi * 8;
    float m[8];
#pragma unroll
    for (int v = 0; v < 8; ++v)
      m[v] = fmaxf(acc[t][v] + b3[rbase + v], 0.f);
#pragma unroll
    for (int mask = 1; mask <= 8; mask <<= 1) {
#pragma unroll
      for (int v = 0; v < 8; ++v)
        m[v] = fmaxf(m[v], __shfl_xor(m[v], mask, 32));
    }
    if (lo == 0) {
#pragma unroll
      for (int v = 0; v < 8; ++v)
        atomicMax(&maxfeat[b*1024 + rbase + v], __float_as_int(m[v]));
    }
  }
}

// ------------------------- decoder FC: out(16,N) = act(in(16,K) @ W^T + b)
__global__ __launch_bounds__(32) void k_fc(
    const float* __restrict__ in, const float* __restrict__ W,
    const float* __restrict__ bias, float* __restrict__ out,
    int K, int N, int relu) {
  const int lane = threadIdx.x & 31;
  const int lo = lane & 15, hi2 = (lane >> 4) << 1;
  const int n0 = blockIdx.x << 4;
  v8f acc = {};
  for (int kk = 0; kk < K; kk += 4) {
    v2f af = *(const v2f*)(in + (size_t)lo * K + kk + hi2);
    v2f bf = *(const v2f*)(W + (size_t)(n0 + lo) * K + kk + hi2);
    acc = __builtin_amdgcn_wmma_f32_16x16x4_f32(
        false, af, false, bf, (short)0, acc, false, false);
  }
  const float bv = bias[n0 + lo];
  const int rb = (lane >> 4) << 3;
#pragma unroll
  for (int v = 0; v < 8; ++v) {
    float o = acc[v] + bv;
    if (relu) o = fmaxf(o, 0.f);
    out[(size_t)(rb + v) * N + n0 + lo] = o;
  }
}

// --------------------------------------------------------------- layernorm
__global__ __launch_bounds__(256) void k_ln(
    const float* __restrict__ h, const float* __restrict__ g,
    const float* __restrict__ bb, float* __restrict__ outpc) {
  __shared__ float s1[256], s2[256];
  const int b = blockIdx.x, tid = threadIdx.x;
  const float* hr = h + (size_t)b * 6144;
  float a = 0.f, aa = 0.f;
  for (int i = tid; i < 6144; i += 256) { float v = hr[i]; a += v; aa += v*v; }
  s1[tid] = a; s2[tid] = aa;
  __syncthreads();
  for (int s = 128; s > 0; s >>= 1) {
    if (tid < s) { s1[tid] += s1[tid+s]; s2[tid] += s2[tid+s]; }
    __syncthreads();
  }
  const float mu  = s1[0] * (1.f/6144.f);
  const float var = s2[0] * (1.f/6144.f) - mu*mu;
  const float rs  = rsqrtf(var + 1e-5f);
  float* orow = outpc + (size_t)b * 6144;
  for (int i = tid; i < 6144; i += 256)
    orow[i] = (hr[i] - mu) * rs * g[i] + bb[i];
}

// ---------------------------------------------------------- |p|^2 and |q|^2
__global__ void k_pq2(const float* __restrict__ pc, const float* __restrict__ qc,
                      float* __restrict__ p2, float* __restrict__ q2) {
  const int i = blockIdx.x * 256 + threadIdx.x;
  if (i < 65536) {
    const float* p = pc + (size_t)i * 3;
    p2[i] = p[0]*p[0] + p[1]*p[1] + p[2]*p[2];
  } else if (i < 98304) {
    const int j = i - 65536;
    const float* q = qc + (size_t)j * 3;
    q2[j] = q[0]*q[0] + q[1]*q[1] + q[2]*q[2];
  }
}

// ------------------------- chamfer: -2*p.q^T via WMMA (K=3 padded to K=4)
__global__ __launch_bounds__(32) void k_chamfer(
    const float* __restrict__ pc, const float* __restrict__ qc,
    const float* __restrict__ p2, const float* __restrict__ q2,
    float* __restrict__ minp, int* __restrict__ minq) {
  const int b  = blockIdx.y;
  const int n0 = blockIdx.x << 4;
  const int lane = threadIdx.x & 31;
  const int lo = lane & 15, hi = lane >> 4, k0 = hi << 1;
  const float* prow = pc + ((size_t)b * N_IN + n0 + lo) * 3;
  v2f af;
  af.x = prow[k0];
  af.y = (k0 == 0) ? prow[1] : 0.f;          // K=3 zero-padded to 4
  float p2r[8], rmin[8];
#pragma unroll
  for (int v = 0; v < 8; ++v) {
    p2r[v]  = p2[b * N_IN + n0 + hi * 8 + v];
    rmin[v] = 3.4e38f;
  }
  for (int j = 0; j < 128; ++j) {
    const float* qrow = qc + ((size_t)b * N_OUT + j*16 + lo) * 3;
    v2f bf;
    bf.x = qrow[k0];
    bf.y = (k0 == 0) ? qrow[1] : 0.f;
    v8f g = {};
    g = __builtin_amdgcn_wmma_f32_16x16x4_f32(
        false, af, false, bf, (short)0, g, false, false);
    const float qq = q2[b * N_OUT + j*16 + lo];
    float cm = 3.4e38f;
#pragma unroll
    for (int v = 0; v < 8; ++v) {
      const float d = p2r[v] + qq - 2.f * g[v];
      rmin[v] = fminf(rmin[v], d);
      cm = fminf(cm, d);
    }
    cm = fminf(cm, __shfl_xor(cm, 16, 32));
    if (lane < 16)
      atomicMin(&minq[b * N_OUT + j*16 + lane], __float_as_int(fmaxf(cm, 0.f)));
  }
#pragma unroll
  for (int mask = 1; mask <= 8; mask <<= 1) {
#pragma unroll
    for (int v = 0; v < 8; ++v)
      rmin[v] = fminf(rmin[v], __shfl_xor(rmin[v], mask, 32));
  }
  if (lo == 0) {
#pragma unroll
    for (int v = 0; v < 8; ++v)
      minp[b * N_IN + n0 + hi * 8 + v] = fmaxf(rmin[v], 0.f);
  }
}

// ----------------------------------------------------------------- loss
__global__ __launch_bounds__(256) void k_loss(
    const float* __restrict__ minp, const int* __restrict__ minq,
    float* __restrict__ loss) {
  __shared__ float s[256];
  const int tid = threadIdx.x;
  float a = 0.f;
  for (int i = tid; i < 65536; i += 256) a += minp[i];
  a *= (1.f/65536.f);
  float c = 0.f;
  for (int i = tid; i < 32768; i += 256) c += __int_as_float(minq[i]);
  a += c * (1.f/32768.f);
  s[tid] = a;
  __syncthreads();
  for (int ss = 128; ss > 0; ss >>= 1) {
    if (tid < ss) s[tid] += s[tid+ss];
    __syncthreads();
  }
  if (tid == 0) loss[0] = s[0];
}

// ================================================================= launch
extern "C" void kernel_launch(void* const* d_in, const int* in_sizes, int n_in,
                              void* d_out, int out_size, void* d_ws, size_t ws_size,
                              hipStream_t stream) {
  const float* in_pc = (const float*)d_in[0];
  const float* w1 = (const float*)d_in[1];  const float* b1 = (const float*)d_in[2];
  const float* w2 = (const float*)d_in[3];  const float* b2 = (const float*)d_in[4];
  const float* w3 = (const float*)d_in[5];  const float* b3 = (const float*)d_in[6];
  const float* we = (const float*)d_in[7];  const float* be = (const float*)d_in[8];
  const float* wf1 = (const float*)d_in[9];  const float* bf1 = (const float*)d_in[10];
  const float* wf2 = (const float*)d_in[11]; const float* bf2 = (const float*)d_in[12];
  const float* wf3 = (const float*)d_in[13]; const float* bf3 = (const float*)d_in[14];
  const float* lng = (const float*)d_in[15]; const float* lnb = (const float*)d_in[16];
  float* out = (float*)d_out;

  // workspace layout (all 8-byte aligned element offsets)
  int*   gfeat   = (int*)d_ws;             // 16*128
  int*   maxfeat = gfeat + 2048;           // 16*1024
  float* h1      = (float*)(maxfeat + 16384); // 16*512
  float* h2      = h1 + 8192;              // 16*1024
  float* h3      = h2 + 16384;             // 16*6144
  float* p2      = h3 + 98304;             // 16*4096
  float* q2      = p2 + 65536;             // 16*2048
  float* minp    = q2 + 32768;             // 16*4096
  int*   minq    = (int*)(minp + 65536);   // 16*2048

  k_init<<<dim3(128), dim3(256), 0, stream>>>(gfeat, maxfeat, minq);
  k_enc12<<<dim3(256), dim3(256), 0, stream>>>(in_pc, w1, b1, w2, b2, gfeat);
  k_conv3<<<dim3(256, 16), dim3(256), 0, stream>>>(in_pc, w1, b1, w2, b2,
                                                   w3, b3, gfeat, out, maxfeat);
  // decoder (M = batch = 16 maps onto one WMMA tile row-block)
  k_fc<<<dim3(16),  dim3(32), 0, stream>>>((const float*)maxfeat, we,  be,
                                           out + EMB_OFF, 1024, 256, 0);
  k_fc<<<dim3(32),  dim3(32), 0, stream>>>(out + EMB_OFF, wf1, bf1, h1, 256, 512, 1);
  k_fc<<<dim3(64),  dim3(32), 0, stream>>>(h1, wf2, bf2, h2, 512, 1024, 1);
  k_fc<<<dim3(384), dim3(32), 0, stream>>>(h2, wf3, bf3, h3, 1024, 6144, 0);
  k_ln<<<dim3(16), dim3(256), 0, stream>>>(h3, lng, lnb, out + OUTPC_OFF);
  k_pq2<<<dim3(384), dim3(256), 0, stream>>>(in_pc, out + OUTPC_OFF, p2, q2);
  k_chamfer<<<dim3(256, 16), dim3(32), 0, stream>>>(in_pc, out + OUTPC_OFF,
                                                    p2, q2, minp, minq);
  k_loss<<<dim3(1), dim3(256), 0, stream>>>(minp, minq, out + LOSS_OFF);
}